// FGNN_51462298140917
// MI455X (gfx1250) — compile-verified
//
#include <hip/hip_runtime.h>
#include <hip/hip_bf16.h>

typedef float v2f __attribute__((ext_vector_type(2)));
typedef float v8f __attribute__((ext_vector_type(8)));

#define Bn 8
#define Nn 1024
#define En 16384
#define Fn 96
#define Hn 12

static __device__ inline v8f wmma4(v2f a, v2f b, v8f c) {
    // V_WMMA_F32_16X16X4_F32 : D = A(16x4) * B(4x16) + C(16x16)
    return __builtin_amdgcn_wmma_f32_16x16x4_f32(
        /*neg_a=*/false, a, /*neg_b=*/false, b,
        /*c_mod=*/(short)0, c, /*reuse_a=*/false, /*reuse_b=*/false);
}

static __device__ inline float gelu_erf(float x) {
    return 0.5f * x * (1.0f + erff(x * 0.7071067811865475f));
}

// ---------------------------------------------------------------------------
// Kernel 1: per-edge message MLP (224->96) + GELU + LayerNorm, weighted
// messages out, atomic scatter-add into aggregated[b, dst, :].
// One wave (32 threads) per 16-edge tile.
// ---------------------------------------------------------------------------
__global__ __launch_bounds__(32) void msg_kernel(
    const float* __restrict__ nodes, const float* __restrict__ efeat,
    const int* __restrict__ edges, const float* __restrict__ ew,
    const float* __restrict__ edrop, const float* __restrict__ Wm,
    const float* __restrict__ bm, const float* __restrict__ gm,
    const float* __restrict__ betam, float* __restrict__ wm_out,
    float* __restrict__ agg)
{
    __shared__ float xin[16][224];
    __shared__ float gout[16][96];
    __shared__ int   sidx[16], didx[16];
    __shared__ float wrow[16], drow[16];
    __shared__ float mu_s[16], rs_s[16];

    const int lane  = threadIdx.x;
    const int tile  = blockIdx.x;                 // B*E/16 tiles
    const int b     = tile / (En / 16);
    const int ebase = (tile % (En / 16)) * 16;

    if (lane < 16) {
        size_t e2 = ((size_t)b * En + ebase + lane) * 2;
        sidx[lane] = edges[e2 + 0];
        didx[lane] = edges[e2 + 1];
        wrow[lane] = ew[(size_t)b * En + ebase + lane];
        drow[lane] = edrop[e2 + 1];
    }
    __syncthreads();

    for (int idx = lane; idx < 16 * 224; idx += 32) {
        int row = idx / 224, c = idx % 224;
        float val;
        if (c < 96)       val = nodes[((size_t)b * Nn + sidx[row]) * 96 + c];
        else if (c < 192) val = nodes[((size_t)b * Nn + didx[row]) * 96 + (c - 96)];
        else              val = efeat[((size_t)b * En + ebase + row) * 32 + (c - 192)];
        xin[row][c] = val;
    }
    __syncthreads();

    const int mrow = lane & 15;
    const int koff = (lane >> 4) << 1;            // 0 or 2

    for (int nt = 0; nt < 6; ++nt) {
        v8f acc = {};
        const int ncol = nt * 16 + mrow;
        for (int ks = 0; ks < 56; ++ks) {
            const int k0 = ks * 4 + koff;
            v2f a = { xin[mrow][k0], xin[mrow][k0 + 1] };
            v2f bf = { Wm[(size_t)k0 * 96 + ncol], Wm[(size_t)(k0 + 1) * 96 + ncol] };
            acc = wmma4(a, bf, acc);
        }
        const float bb = bm[ncol];
        for (int i = 0; i < 8; ++i) {
            int row = i + ((lane >> 4) << 3);
            gout[row][ncol] = gelu_erf(acc[i] + bb);
        }
    }
    __syncthreads();

    if (lane < 16) {
        float s = 0.f;
        for (int n = 0; n < 96; ++n) s += gout[lane][n];
        float mu = s * (1.0f / 96.0f);
        float var = 0.f;
        for (int n = 0; n < 96; ++n) { float t = gout[lane][n] - mu; var += t * t; }
        mu_s[lane] = mu;
        rs_s[lane] = rsqrtf(var * (1.0f / 96.0f) + 1e-3f);
    }
    __syncthreads();

    for (int idx = lane; idx < 16 * 96; idx += 32) {
        int row = idx / 96, n = idx % 96;
        float val = (gout[row][n] - mu_s[row]) * rs_s[row] * gm[n] + betam[n];
        float wmv = val * wrow[row];
        wm_out[((size_t)b * En + ebase + row) * 96 + n] = wmv;
        atomicAdd(&agg[((size_t)b * Nn + didx[row]) * 96 + n], wmv * drow[row]);
    }
}

// ---------------------------------------------------------------------------
// Kernel 2: Q,K,V,Gate projections (192 -> 96) over [nodes | aggregated].
// ---------------------------------------------------------------------------
__global__ __launch_bounds__(32) void qkvg_kernel(
    const float* __restrict__ nodes, const float* __restrict__ agg,
    const float* __restrict__ Wq, const float* __restrict__ bq,
    const float* __restrict__ Wk, const float* __restrict__ bk,
    const float* __restrict__ Wv, const float* __restrict__ bv,
    const float* __restrict__ Wg, const float* __restrict__ bg,
    float* __restrict__ qo, float* __restrict__ ko,
    float* __restrict__ vo, float* __restrict__ go)
{
    __shared__ float xin[16][192];
    const int lane  = threadIdx.x;
    const int tile  = blockIdx.x;                 // B*N/16 tiles
    const int b     = tile / (Nn / 16);
    const int nbase = (tile % (Nn / 16)) * 16;

    for (int idx = lane; idx < 16 * 192; idx += 32) {
        int row = idx / 192, c = idx % 192;
        xin[row][c] = (c < 96) ? nodes[((size_t)b * Nn + nbase + row) * 96 + c]
                               : agg[((size_t)b * Nn + nbase + row) * 96 + (c - 96)];
    }
    __syncthreads();

    const float* Ws[4]   = { Wq, Wk, Wv, Wg };
    const float* bsv[4]  = { bq, bk, bv, bg };
    float*       outs[4] = { qo, ko, vo, go };

    const int mrow = lane & 15;
    const int koff = (lane >> 4) << 1;

    for (int m = 0; m < 4; ++m) {
        const float* W = Ws[m];
        const float* bias = bsv[m];
        float* out = outs[m];
        for (int nt = 0; nt < 6; ++nt) {
            v8f acc = {};
            const int ncol = nt * 16 + mrow;
            for (int ks = 0; ks < 48; ++ks) {
                const int k0 = ks * 4 + koff;
                v2f a = { xin[mrow][k0], xin[mrow][k0 + 1] };
                v2f bf = { W[(size_t)k0 * 96 + ncol], W[(size_t)(k0 + 1) * 96 + ncol] };
                acc = wmma4(a, bf, acc);
            }
            const float bb = bias[ncol];
            for (int i = 0; i < 8; ++i) {
                int row = i + ((lane >> 4) << 3);
                float val = acc[i] + bb;
                if (m == 3) val = 1.0f / (1.0f + expf(-val));   // sigmoid gate
                out[((size_t)b * Nn + nbase + row) * 96 + ncol] = val;
            }
        }
    }
}

// ---------------------------------------------------------------------------
// Kernel 3: gated MHSA, one wave per (b, h, 16-query tile). Two-pass softmax:
// pass A accumulates row max/sum (S recomputed in pass B). S = Q*K^T via WMMA,
// P*V via WMMA (only 8 valid N columns of d=8).
// ---------------------------------------------------------------------------
__global__ __launch_bounds__(32) void attn_kernel(
    const float* __restrict__ qm, const float* __restrict__ km,
    const float* __restrict__ vmat, const float* __restrict__ gm,
    float* __restrict__ att)
{
    __shared__ float qt[16][8];
    __shared__ float st[16][16];
    __shared__ float rowm[16], rowl[16];

    const int lane  = threadIdx.x;
    const int id    = blockIdx.x;                 // b*H*64 + h*64 + qtile
    const int qtile = id & 63;
    const int h     = (id >> 6) % Hn;
    const int b     = id / (64 * Hn);
    const int qbase = qtile * 16;
    const float scale = 0.3535533905932738f;      // 1/sqrt(8)

    for (int idx = lane; idx < 128; idx += 32) {
        int row = idx >> 3, c = idx & 7;
        qt[row][c] = qm[((size_t)b * Nn + qbase + row) * 96 + h * 8 + c] * scale;
    }
    __syncthreads();

    const int mrow = lane & 15;
    const int koff = (lane >> 4) << 1;

    float m_i = -3.0e38f, l_i = 0.f;

    // ---- pass A: softmax statistics ----
    for (int kt = 0; kt < 64; ++kt) {
        v8f sacc = {};
        for (int s = 0; s < 2; ++s) {
            const int k0 = s * 4 + koff;
            v2f a = { qt[mrow][k0], qt[mrow][k0 + 1] };
            const float* kp = &km[((size_t)b * Nn + kt * 16 + mrow) * 96 + h * 8];
            v2f bf = { kp[k0], kp[k0 + 1] };
            sacc = wmma4(a, bf, sacc);
        }
        for (int i = 0; i < 8; ++i) {
            int row = i + ((lane >> 4) << 3);
            st[row][mrow] = sacc[i];
        }
        __syncthreads();
        if (lane < 16) {
            float mx = m_i;
            for (int j = 0; j < 16; ++j) mx = fmaxf(mx, st[lane][j]);
            float l = l_i * expf(m_i - mx);
            for (int j = 0; j < 16; ++j) l += expf(st[lane][j] - mx);
            m_i = mx; l_i = l;
        }
        __syncthreads();
    }
    if (lane < 16) { rowm[lane] = m_i; rowl[lane] = l_i; }
    __syncthreads();

    // ---- pass B: O = softmax(S) @ V ----
    v8f oacc = {};
    for (int kt = 0; kt < 64; ++kt) {
        v8f sacc = {};
        for (int s = 0; s < 2; ++s) {
            const int k0 = s * 4 + koff;
            v2f a = { qt[mrow][k0], qt[mrow][k0 + 1] };
            const float* kp = &km[((size_t)b * Nn + kt * 16 + mrow) * 96 + h * 8];
            v2f bf = { kp[k0], kp[k0 + 1] };
            sacc = wmma4(a, bf, sacc);
        }
        for (int i = 0; i < 8; ++i) {
            int row = i + ((lane >> 4) << 3);
            st[row][mrow] = expf(sacc[i] - rowm[row]) / rowl[row];
        }
        __syncthreads();
        for (int s = 0; s < 4; ++s) {
            const int k0 = s * 4 + koff;
            v2f a = { st[mrow][k0], st[mrow][k0 + 1] };     // P tile, M=q row, K=local col
            const int node0 = kt * 16 + k0;
            float b0 = (mrow < 8) ? vmat[((size_t)b * Nn + node0)     * 96 + h * 8 + mrow] : 0.f;
            float b1 = (mrow < 8) ? vmat[((size_t)b * Nn + node0 + 1) * 96 + h * 8 + mrow] : 0.f;
            v2f bf = { b0, b1 };
            oacc = wmma4(a, bf, oacc);
        }
        __syncthreads();
    }

    for (int i = 0; i < 8; ++i) {
        int row = i + ((lane >> 4) << 3);
        if (mrow < 8) {
            size_t off = ((size_t)b * Nn + qbase + row) * 96 + h * 8 + mrow;
            att[off] = oacc[i] * gm[off];          // gate already sigmoided
        }
    }
}

// ---------------------------------------------------------------------------
// Kernel 4: output projection (96 -> 96) + GELU + LayerNorm.
// ---------------------------------------------------------------------------
__global__ __launch_bounds__(32) void out_kernel(
    const float* __restrict__ att, const float* __restrict__ Wo,
    const float* __restrict__ bo, const float* __restrict__ gu,
    const float* __restrict__ betau, float* __restrict__ upd)
{
    __shared__ float xin[16][96];
    __shared__ float gout[16][96];
    __shared__ float mu_s[16], rs_s[16];

    const int lane  = threadIdx.x;
    const int tile  = blockIdx.x;                 // B*N/16 tiles
    const int b     = tile / (Nn / 16);
    const int nbase = (tile % (Nn / 16)) * 16;

    for (int idx = lane; idx < 16 * 96; idx += 32) {
        int row = idx / 96, c = idx % 96;
        xin[row][c] = att[((size_t)b * Nn + nbase + row) * 96 + c];
    }
    __syncthreads();

    const int mrow = lane & 15;
    const int koff = (lane >> 4) << 1;

    for (int nt = 0; nt < 6; ++nt) {
        v8f acc = {};
        const int ncol = nt * 16 + mrow;
        for (int ks = 0; ks < 24; ++ks) {
            const int k0 = ks * 4 + koff;
            v2f a = { xin[mrow][k0], xin[mrow][k0 + 1] };
            v2f bf = { Wo[(size_t)k0 * 96 + ncol], Wo[(size_t)(k0 + 1) * 96 + ncol] };
            acc = wmma4(a, bf, acc);
        }
        const float bb = bo[ncol];
        for (int i = 0; i < 8; ++i) {
            int row = i + ((lane >> 4) << 3);
            gout[row][ncol] = gelu_erf(acc[i] + bb);
        }
    }
    __syncthreads();

    if (lane < 16) {
        float s = 0.f;
        for (int n = 0; n < 96; ++n) s += gout[lane][n];
        float mu = s * (1.0f / 96.0f);
        float var = 0.f;
        for (int n = 0; n < 96; ++n) { float t = gout[lane][n] - mu; var += t * t; }
        mu_s[lane] = mu;
        rs_s[lane] = rsqrtf(var * (1.0f / 96.0f) + 1e-3f);
    }
    __syncthreads();

    for (int idx = lane; idx < 16 * 96; idx += 32) {
        int row = idx / 96, n = idx % 96;
        upd[((size_t)b * Nn + nbase + row) * 96 + n] =
            (gout[row][n] - mu_s[row]) * rs_s[row] * gu[n] + betau[n];
    }
}

// ---------------------------------------------------------------------------
// Kernel 5: pass-through tuple tail: edges (int->float), edge_weights,
// edge_dropout.
// ---------------------------------------------------------------------------
__global__ void tail_kernel(const int* __restrict__ edges,
                            const float* __restrict__ ew,
                            const float* __restrict__ ed,
                            float* __restrict__ out)
{
    const int nE2 = Bn * En * 2;
    const int nEw = Bn * En;
    int i = blockIdx.x * blockDim.x + threadIdx.x;
    if (i < nE2)                  out[i] = (float)edges[i];
    else if (i < nE2 + nEw)       out[i] = ew[i - nE2];
    else if (i < nE2 + nEw + nE2) out[i] = ed[i - nE2 - nEw];
}

// ---------------------------------------------------------------------------
extern "C" void kernel_launch(void* const* d_in, const int* in_sizes, int n_in,
                              void* d_out, int out_size, void* d_ws, size_t ws_size,
                              hipStream_t stream) {
    const float* nodes = (const float*)d_in[0];
    const float* efeat = (const float*)d_in[1];
    const int*   edges = (const int*)d_in[2];
    const float* ew    = (const float*)d_in[3];
    const float* ed    = (const float*)d_in[4];
    const float* Wm    = (const float*)d_in[5];
    const float* bm    = (const float*)d_in[6];
    const float* gmm   = (const float*)d_in[7];
    const float* betam = (const float*)d_in[8];
    const float* Wq    = (const float*)d_in[9];
    const float* bq    = (const float*)d_in[10];
    const float* Wk    = (const float*)d_in[11];
    const float* bk    = (const float*)d_in[12];
    const float* Wv    = (const float*)d_in[13];
    const float* bv    = (const float*)d_in[14];
    const float* Wg    = (const float*)d_in[15];
    const float* bg    = (const float*)d_in[16];
    const float* Wo    = (const float*)d_in[17];
    const float* bo    = (const float*)d_in[18];
    const float* gu    = (const float*)d_in[19];
    const float* betau = (const float*)d_in[20];

    // output layout (floats): updated_nodes | weighted_messages | edges | ew | ed
    const size_t SZ_UPD = (size_t)Bn * Nn * Fn;          // 786432
    const size_t SZ_WM  = (size_t)Bn * En * Fn;          // 12582912
    const size_t SZ_E2  = (size_t)Bn * En * 2;           // 262144
    float* out    = (float*)d_out;
    float* upd    = out;
    float* wm_out = out + SZ_UPD;
    float* tail   = out + SZ_UPD + SZ_WM;

    // workspace layout (floats): agg | q | k | v | gate | att  (6 * 786432)
    float* ws   = (float*)d_ws;
    float* agg  = ws;
    float* qo   = ws + 1 * SZ_UPD;
    float* ko   = ws + 2 * SZ_UPD;
    float* vo   = ws + 3 * SZ_UPD;
    float* go   = ws + 4 * SZ_UPD;
    float* atto = ws + 5 * SZ_UPD;

    hipMemsetAsync(agg, 0, SZ_UPD * sizeof(float), stream);

    msg_kernel<<<(Bn * En) / 16, 32, 0, stream>>>(
        nodes, efeat, edges, ew, ed, Wm, bm, gmm, betam, wm_out, agg);

    qkvg_kernel<<<(Bn * Nn) / 16, 32, 0, stream>>>(
        nodes, agg, Wq, bq, Wk, bk, Wv, bv, Wg, bg, qo, ko, vo, go);

    attn_kernel<<<Bn * Hn * (Nn / 16), 32, 0, stream>>>(qo, ko, vo, go, atto);

    out_kernel<<<(Bn * Nn) / 16, 32, 0, stream>>>(atto, Wo, bo, gu, betau, upd);

    tail_kernel<<<((2 * SZ_E2 + Bn * En) + 255) / 256, 256, 0, stream>>>(
        edges, ew, ed, tail);
}